// RCOptimModel_51290499449175
// MI455X (gfx1250) — compile-verified
//
#include <hip/hip_runtime.h>
#include <math.h>

// ---------------- problem constants ----------------
#define NS 130              // state dim (2 + 128 rooms)
#define NROOMS 128
#define TSTEPS 131072       // output rows
#define SSTEPS 131071       // RK4 steps (T-1)
#define DTs 30.0f
#define DAYs 86400.0f
#define TOUTN 1100
#define TWO_PI 6.28318530717958647692f

// scan decomposition: C chunks of L steps
#define CHUNK_L 256
#define NCHUNK 512          // 512*256 = 131072 >= SSTEPS; last chunk short

// forcing GEMM geometry
#define WSTRIDE 144         // padded row stride of W (9 x 16)
#define KTOT 384            // K = 128(Qf@t)+128(Qf@t+h)+128(Qf@t+dt lifted)
#define KC 32               // K chunk staged in LDS
#define BT 128              // time rows per block (8 waves x 16)
#define QPITCH 36           // Q panel pitch (8B aligned frags, bank-disjoint halves)
#define MPITCH 132          // padded M row pitch in pass kernels (16B aligned)

typedef float v2f __attribute__((ext_vector_type(2)));
typedef float v8f __attribute__((ext_vector_type(8)));

__device__ __forceinline__ float sigm(float x) { return 1.0f / (1.0f + __expf(-x)); }

// jnp.interp on uniform 3600s grid (t always inside [0, (TOUTN-1)*3600) here)
__device__ __forceinline__ float toutInterp(float t, const float* __restrict__ tv) {
    float u = t * (1.0f / 3600.0f);
    int i = (int)floorf(u);
    i = max(0, min(i, TOUTN - 2));
    float fr = u - (float)i;
    return fmaf(fr, tv[i + 1] - tv[i], tv[i]);
}

// ---------------- setup: A matrix from params ----------------
__global__ void k_setupA(const float* __restrict__ params, float* __restrict__ A) {
    int idx = blockIdx.x * blockDim.x + threadIdx.x;
    if (idx >= NS * NS) return;
    int i = idx / NS, j = idx % NS;
    if (i != j) {
        A[idx] = sigm(params[idx]) * 1e-5f;
    } else {
        float s = 0.0f;
        for (int jj = 0; jj < NS; ++jj) s += sigm(params[i * NS + jj]);
        A[idx] = -s * 1e-5f;   // A[i][i] = -sum_j K[i][j]  (incl diag)
    }
}

// ---------------- generic 130x130 matmul (L2-hot, tiny) ----------------
__global__ void k_matmul(const float* __restrict__ Aa, const float* __restrict__ Bb,
                         float* __restrict__ Cc) {
    int idx = blockIdx.x * blockDim.x + threadIdx.x;
    if (idx >= NS * NS) return;
    int i = idx / NS, j = idx % NS;
    float s = 0.0f;
    for (int k = 0; k < NS; ++k) s = fmaf(Aa[i * NS + k], Bb[k * NS + j], s);
    Cc[idx] = s;
}

// ---------------- combine: M = I+dtA+..., Bmat = forcing GEMM B-panel ----------------
__global__ void k_combine(const float* __restrict__ A, const float* __restrict__ A2,
                          const float* __restrict__ A3, const float* __restrict__ A4,
                          float* __restrict__ M, float* __restrict__ Bmat) {
    const float h2 = DTs * DTs * 0.5f;
    const float h3 = DTs * DTs * DTs / 6.0f;
    const float h4 = DTs * DTs * DTs * DTs / 24.0f;
    const float h3q = DTs * DTs * DTs * 0.25f;
    const float cB = (DTs / 6.0f) * 1e-6f;
    int idx = blockIdx.x * blockDim.x + threadIdx.x;
    if (idx < NS * NS) {
        int i = idx / NS, j = idx % NS;
        float d = (i == j) ? 1.0f : 0.0f;
        M[idx] = d + DTs * A[idx] + h2 * A2[idx] + h3 * A3[idx] + h4 * A4[idx];
    } else {
        int e = idx - NS * NS;
        if (e >= KTOT * WSTRIDE) return;
        int k = e / WSTRIDE, col = e % WSTRIDE;
        float v = 0.0f;
        if (col < NS) {
            if (k < 128) {                       // (dt/6)*1e-6 * P1[col][k+2]
                int j = k + 2, m = col * NS + j;
                float d = (col == j) ? 1.0f : 0.0f;
                v = cB * (d + DTs * A[m] + h2 * A2[m] + h3q * A3[m]);
            } else if (k < 256) {                // (dt/6)*1e-6 * P2[col][k-126]
                int j = k - 126, m = col * NS + j;
                float d = (col == j) ? 4.0f : 0.0f;
                v = cB * (d + 2.0f * DTs * A[m] + h2 * A2[m]);
            } else {                             // lifted identity (g at t+dt)
                v = (col == (k - 254)) ? cB : 0.0f;
            }
        }
        Bmat[e] = v;
    }
}

// ---------------- pv vectors: (dt/6)*P1*b0, (dt/6)*P2*b0, (dt/6)*b0 ----------------
__global__ void k_vectors(const float* __restrict__ A, const float* __restrict__ A2,
                          const float* __restrict__ A3, const float* __restrict__ Bin,
                          float* __restrict__ pv) {
    __shared__ float b0[NS];
    int t = threadIdx.x;
    if (t < NS) b0[t] = Bin[t * (1 + NROOMS)];   // B[:,0]
    __syncthreads();
    if (t < NS) {
        float s1 = 0.0f, s2 = 0.0f, s3 = 0.0f;
        for (int j = 0; j < NS; ++j) {
            float b = b0[j];
            s1 = fmaf(A[t * NS + j], b, s1);
            s2 = fmaf(A2[t * NS + j], b, s2);
            s3 = fmaf(A3[t * NS + j], b, s3);
        }
        const float c = DTs / 6.0f;
        const float h2 = DTs * DTs * 0.5f;
        const float h3q = DTs * DTs * DTs * 0.25f;
        pv[t]          = c * (b0[t] + DTs * s1 + h2 * s2 + h3q * s3);
        pv[NS + t]     = c * (4.0f * b0[t] + 2.0f * DTs * s1 + h2 * s2);
        pv[2 * NS + t] = c * b0[t];
    }
}

// ---------------- forcing table W via WMMA GEMM ----------------
// W[n][0..129] = (dt/6)(P1 g(t_n) + P2 g(t_n+h) + g(t_n+dt)), n in [0, SSTEPS)
// GEMM part: Q(time x 384) @ Bmat(384 x 144); Tout rank-1 terms added in epilogue.
__global__ __launch_bounds__(256) void k_forcing(const float* __restrict__ heat,
                                                 const float* __restrict__ toutv,
                                                 const float* __restrict__ Bmat,
                                                 const float* __restrict__ pv,
                                                 float* __restrict__ W) {
    __shared__ __align__(16) float Qp[BT][QPITCH];   // A-panel stage (b64 frags)
    __shared__ __align__(16) v2f BpP[KC / 2][WSTRIDE]; // B-panel, K-pair interleaved
    __shared__ float qa_s[NROOMS], ph_s[NROOMS], sh_s[NROOMS];
    __shared__ float Tg_s[BT + 1], Tm_s[BT];
    __shared__ float pv_s[3 * NS];

    const int tid = threadIdx.x;
    const int n0 = blockIdx.x * BT;

    if (tid < NROOMS) {
        qa_s[tid] = 500.0f * sigm(heat[3 * tid + 0]);
        ph_s[tid] = heat[3 * tid + 1];
        sh_s[tid] = heat[3 * tid + 2];
    }
    for (int e = tid; e < 3 * NS; e += 256) pv_s[e] = pv[e];
    if (tid < BT + 1) Tg_s[tid] = toutInterp((float)(n0 + tid) * DTs, toutv);
    if (tid >= 128 && tid < 256) Tm_s[tid - 128] = toutInterp((float)(n0 + tid - 128) * DTs + 15.0f, toutv);

    const int lane = tid & 31, wave = tid >> 5;
    const int hi = lane >> 4, lo = lane & 15;
    const int tw = wave * 16;                 // this wave's 16 time rows

    v8f acc[9];
#pragma unroll
    for (int ct = 0; ct < 9; ++ct) acc[ct] = (v8f){0.f, 0.f, 0.f, 0.f, 0.f, 0.f, 0.f, 0.f};

    for (int kc = 0; kc < KTOT; kc += KC) {
        __syncthreads();
        // stage Q tile: Qf evaluated on the fly (sin + sigmoid)
        for (int e = tid; e < BT * KC; e += 256) {
            int tt = e >> 5;           // /KC
            int kk = e & (KC - 1);
            int k = kc + kk;
            int room = k & 127;
            float toff = (k < 128) ? 0.0f : ((k < 256) ? 15.0f : 30.0f);
            float t = (float)(n0 + tt) * DTs + toff;
            float s = __sinf(TWO_PI * t * (1.0f / DAYs) + ph_s[room]);
            Qp[tt][kk] = qa_s[room] * sigm(sh_s[room] * s);
        }
        // stage B tile, K-pair interleaved so each fragment is one aligned b64 load
        for (int e = tid; e < (KC / 2) * WSTRIDE; e += 256) {
            int kk2 = e / WSTRIDE, col = e % WSTRIDE;
            v2f b;
            b.x = Bmat[(size_t)(kc + 2 * kk2 + 0) * WSTRIDE + col];
            b.y = Bmat[(size_t)(kc + 2 * kk2 + 1) * WSTRIDE + col];
            BpP[kk2][col] = b;
        }
        __syncthreads();

#pragma unroll
        for (int kk = 0; kk < KC; kk += 4) {
            // A frag (16x4 f32): vgpr v holds K = 2*hi + v for row m = lo
            v2f a = *(const v2f*)&Qp[tw + lo][kk + 2 * hi];
#pragma unroll
            for (int ct = 0; ct < 9; ++ct) {
                // B frag (4x16 f32): vgpr v holds row K = 2*hi + v, col = lo
                v2f b = BpP[(kk >> 1) + hi][ct * 16 + lo];
                acc[ct] = __builtin_amdgcn_wmma_f32_16x16x4_f32(
                    false, a, false, b, (short)0, acc[ct], false, false);
            }
        }
    }

    // epilogue: + Tout rank-1 terms, store (D layout: time = v+8*hi, col = lo)
#pragma unroll
    for (int ct = 0; ct < 9; ++ct) {
        int col = ct * 16 + lo;
        float p1 = (col < NS) ? pv_s[col] : 0.0f;
        float p2 = (col < NS) ? pv_s[NS + col] : 0.0f;
        float p3 = (col < NS) ? pv_s[2 * NS + col] : 0.0f;
#pragma unroll
        for (int v = 0; v < 8; ++v) {
            int tt = tw + v + 8 * hi;
            int n = n0 + tt;
            float val = acc[ct][v] + Tg_s[tt] * p1 + Tm_s[tt] * p2 + Tg_s[tt + 1] * p3;
            if (n < SSTEPS && col < NS) W[(size_t)n * WSTRIDE + col] = val;
        }
    }
}

// ---------------- LDS-resident matvec scan kernels ----------------
// One barrier per step; next forcing vector kept in a register (only its own
// row's thread ever reads it); prefetch (global_prefetch_b8) runs 8 steps ahead.

__device__ __forceinline__ float dotRow(const float* __restrict__ Ms,
                                        const float* __restrict__ xv,
                                        int row, float s) {
    const float4* m4 = (const float4*)&Ms[row * MPITCH];
    const float4* x4 = (const float4*)xv;
#pragma unroll 8
    for (int q = 0; q < 32; ++q) {
        float4 m = m4[q], x = x4[q];
        s = fmaf(m.x, x.x, s); s = fmaf(m.y, x.y, s);
        s = fmaf(m.z, x.z, s); s = fmaf(m.w, x.w, s);
    }
    s = fmaf(Ms[row * MPITCH + 128], xv[128], s);
    s = fmaf(Ms[row * MPITCH + 129], xv[129], s);
    return s;
}

// Pass 1: per-chunk forcing summary r_c = sum_j M^{L-1-j} w_j  (Horner)
__global__ __launch_bounds__(256) void k_pass1(const float* __restrict__ Mg,
                                               const float* __restrict__ W,
                                               float* __restrict__ R) {
    __shared__ __align__(16) float Ms[NS * MPITCH];
    __shared__ __align__(16) float xs[2][MPITCH];
    const int tid = threadIdx.x;
    for (int e = tid; e < NS * NS; e += 256) Ms[(e / NS) * MPITCH + (e % NS)] = Mg[e];
    if (tid < NS) xs[0][tid] = 0.0f;
    const size_t base = (size_t)blockIdx.x * CHUNK_L;
    float wreg = (tid < NS) ? W[base * WSTRIDE + tid] : 0.0f;
    __syncthreads();
    int cur = 0;
    for (int j = 0; j < CHUNK_L; ++j) {
        float nxt = 0.0f;
        if (tid < NS) {
            if (j + 1 < CHUNK_L) nxt = W[(base + j + 1) * WSTRIDE + tid];
            if (j + 8 < CHUNK_L) __builtin_prefetch(&W[(base + j + 8) * WSTRIDE + tid], 0, 1);
            float s = dotRow(Ms, xs[cur], tid, wreg);
            xs[cur ^ 1][tid] = s;
        }
        wreg = nxt;
        cur ^= 1;
        __syncthreads();
    }
    if (tid < NS) R[blockIdx.x * NS + tid] = xs[cur][tid];
}

// Pass 2: sequential chunk-boundary recurrence x_{c+1} = M^L x_c + r_c
__global__ __launch_bounds__(256) void k_pass2(const float* __restrict__ MLg,
                                               const float* __restrict__ R,
                                               const float* __restrict__ toutv,
                                               float* __restrict__ Xc) {
    __shared__ __align__(16) float Ms[NS * MPITCH];
    __shared__ __align__(16) float xs[2][MPITCH];
    const int tid = threadIdx.x;
    for (int e = tid; e < NS * NS; e += 256) Ms[(e / NS) * MPITCH + (e % NS)] = MLg[e];
    const float iv = toutv[0] + 4.0f;
    if (tid < NS) { xs[0][tid] = iv; Xc[tid] = iv; }
    float wreg = (tid < NS) ? R[tid] : 0.0f;
    __syncthreads();
    int cur = 0;
    for (int c = 0; c < NCHUNK - 1; ++c) {
        float nxt = 0.0f;
        if (tid < NS) {
            if (c + 1 < NCHUNK - 1) nxt = R[(c + 1) * NS + tid];
            float s = dotRow(Ms, xs[cur], tid, wreg);
            xs[cur ^ 1][tid] = s;
            Xc[(c + 1) * NS + tid] = s;
        }
        wreg = nxt;
        cur ^= 1;
        __syncthreads();
    }
}

// Pass 3: replay chunk from Xc, emit rooms to out (rows cL .. cL+255 only)
__global__ __launch_bounds__(256) void k_pass3(const float* __restrict__ Mg,
                                               const float* __restrict__ W,
                                               const float* __restrict__ Xc,
                                               float* __restrict__ out) {
    __shared__ __align__(16) float Ms[NS * MPITCH];
    __shared__ __align__(16) float xs[2][MPITCH];
    const int tid = threadIdx.x;
    const int c = blockIdx.x;
    for (int e = tid; e < NS * NS; e += 256) Ms[(e / NS) * MPITCH + (e % NS)] = Mg[e];
    if (tid < NS) xs[0][tid] = Xc[c * NS + tid];
    const size_t rowBase = (size_t)c * CHUNK_L;
    float wreg = (tid < NS) ? W[rowBase * WSTRIDE + tid] : 0.0f;
    __syncthreads();
    if (tid >= 2 && tid < NS) out[rowBase * NROOMS + (tid - 2)] = xs[0][tid];
    int cur = 0;
    for (int j = 0; j < CHUNK_L - 1; ++j) {     // 255 steps; next chunk owns row (c+1)L
        float nxt = 0.0f;
        if (tid < NS) {
            if (j + 1 < CHUNK_L - 1) nxt = W[(rowBase + j + 1) * WSTRIDE + tid];
            if (j + 8 < CHUNK_L - 1) __builtin_prefetch(&W[(rowBase + j + 8) * WSTRIDE + tid], 0, 1);
            float s = dotRow(Ms, xs[cur], tid, wreg);
            xs[cur ^ 1][tid] = s;
            if (tid >= 2) out[(rowBase + j + 1) * NROOMS + (tid - 2)] = s;
        }
        wreg = nxt;
        cur ^= 1;
        __syncthreads();
    }
}

// ---------------- host-side orchestration ----------------
extern "C" void kernel_launch(void* const* d_in, const int* in_sizes, int n_in,
                              void* d_out, int out_size, void* d_ws, size_t ws_size,
                              hipStream_t stream) {
    const float* params  = (const float*)d_in[1];
    const float* heating = (const float*)d_in[2];
    const float* Bin     = (const float*)d_in[3];
    const float* tout_v  = (const float*)d_in[5];
    float* out = (float*)d_out;
    float* ws = (float*)d_ws;

    // workspace layout (floats)
    const size_t MM = (size_t)NS * NS;            // 16900
    float* A    = ws;
    float* A2   = A + MM;
    float* A3   = A2 + MM;
    float* A4   = A3 + MM;
    float* M    = A4 + MM;
    float* MPa  = M + MM;
    float* MPb  = MPa + MM;
    float* Bmat = MPb + MM;                       // 384*144
    float* pv   = Bmat + (size_t)KTOT * WSTRIDE;  // 3*130
    float* R    = pv + 3 * NS;                    // (NCHUNK-1)*130
    float* Xc   = R + (size_t)(NCHUNK - 1) * NS;  // NCHUNK*130
    float* W    = Xc + (size_t)NCHUNK * NS;       // SSTEPS*144  (~75.5 MB)
    size_t need = (size_t)(W - ws) + (size_t)SSTEPS * WSTRIDE;
    if (ws_size < need * sizeof(float)) return;   // workspace too small; bail deterministically

    const int mmBlocks = (int)((MM + 255) / 256);

    // 1) A from params; powers A^2..A^4
    k_setupA<<<mmBlocks, 256, 0, stream>>>(params, A);
    k_matmul<<<mmBlocks, 256, 0, stream>>>(A, A, A2);
    k_matmul<<<mmBlocks, 256, 0, stream>>>(A2, A, A3);
    k_matmul<<<mmBlocks, 256, 0, stream>>>(A3, A, A4);

    // 2) M, B-panel, pv vectors
    const int cmbBlocks = (int)((MM + (size_t)KTOT * WSTRIDE + 255) / 256);
    k_combine<<<cmbBlocks, 256, 0, stream>>>(A, A2, A3, A4, M, Bmat);
    k_vectors<<<1, 160, 0, stream>>>(A, A2, A3, Bin, pv);

    // 3) M^256 by repeated squaring (ends in MPb)
    k_matmul<<<mmBlocks, 256, 0, stream>>>(M, M, MPa);      // M^2
    k_matmul<<<mmBlocks, 256, 0, stream>>>(MPa, MPa, MPb);  // M^4
    k_matmul<<<mmBlocks, 256, 0, stream>>>(MPb, MPb, MPa);  // M^8
    k_matmul<<<mmBlocks, 256, 0, stream>>>(MPa, MPa, MPb);  // M^16
    k_matmul<<<mmBlocks, 256, 0, stream>>>(MPb, MPb, MPa);  // M^32
    k_matmul<<<mmBlocks, 256, 0, stream>>>(MPa, MPa, MPb);  // M^64
    k_matmul<<<mmBlocks, 256, 0, stream>>>(MPb, MPb, MPa);  // M^128
    k_matmul<<<mmBlocks, 256, 0, stream>>>(MPa, MPa, MPb);  // M^256

    // 4) forcing table via WMMA GEMM (1024 blocks x 128 time rows)
    k_forcing<<<TSTEPS / BT, 256, 0, stream>>>(heating, tout_v, Bmat, pv, W);

    // 5) affine scan: chunk summaries -> boundary states -> replay
    k_pass1<<<NCHUNK - 1, 256, 0, stream>>>(M, W, R);
    k_pass2<<<1, 256, 0, stream>>>(MPb, R, tout_v, Xc);
    k_pass3<<<NCHUNK, 256, 0, stream>>>(M, W, Xc, out);
    (void)in_sizes; (void)n_in; (void)out_size;
}